// EmergencyBraking_8993661517959
// MI455X (gfx1250) — compile-verified
//
#include <hip/hip_runtime.h>

typedef __attribute__((ext_vector_type(2))) float v2f;
typedef __attribute__((ext_vector_type(4))) float v4f;
typedef __attribute__((ext_vector_type(8))) float v8f;

#define NPIX  15000
#define NPIX4 3750
#define IMG_H 200
#define IMG_W 75

// ---------------------------------------------------------------------------
// Kernel 0: hoist the transpose permutation of the frozen lane weights.
// w_perm[p] = laneW[(p % 200) * 75 + (p / 200)]  so the streaming kernel
// reads both operands fully contiguously.
// ---------------------------------------------------------------------------
__global__ void permute_lane_w(const float* __restrict__ laneW,
                               float* __restrict__ wperm) {
    int p = blockIdx.x * blockDim.x + threadIdx.x;
    if (p < NPIX) {
        int i = p % IMG_H;
        int j = p / IMG_H;
        wperm[p] = laneW[i * IMG_W + j];
    }
}

// ---------------------------------------------------------------------------
// Kernel 1: HBM-bound part.  steer[b] = sum_p (rgb[b,p] > 0) * wperm[p] + lane_b
// One 256-thread block per batch row, b128 nontemporal streaming loads
// (491 MB total read; ~21 us floor at 23.3 TB/s).
// ---------------------------------------------------------------------------
__global__ void __launch_bounds__(256)
steer_kernel(const float* __restrict__ rgb,
             const float* __restrict__ wperm,
             const float* __restrict__ lane_b,
             float* __restrict__ steer) {
    const int b = blockIdx.x;
    const v4f* rgb4 = reinterpret_cast<const v4f*>(rgb + (size_t)b * NPIX);
    const v4f* w4   = reinterpret_cast<const v4f*>(wperm);

    float sum = 0.f;
    for (int idx = threadIdx.x; idx < NPIX4; idx += 256) {
        v4f x = __builtin_nontemporal_load(&rgb4[idx]);   // streamed once
        v4f w = w4[idx];                                  // L2-resident, reused
        sum += (x.x > 0.f ? w.x : 0.f);
        sum += (x.y > 0.f ? w.y : 0.f);
        sum += (x.z > 0.f ? w.z : 0.f);
        sum += (x.w > 0.f ? w.w : 0.f);
    }
    // wave32 reduction
    #pragma unroll
    for (int off = 16; off > 0; off >>= 1)
        sum += __shfl_down(sum, off, 32);

    __shared__ float part[8];
    const int lane = threadIdx.x & 31;
    const int wv   = threadIdx.x >> 5;
    if (lane == 0) part[wv] = sum;
    __syncthreads();
    if (threadIdx.x == 0) {
        float t = 0.f;
        #pragma unroll
        for (int i = 0; i < 8; ++i) t += part[i];
        steer[b] = t + lane_b[0];
    }
}

// ---------------------------------------------------------------------------
// Kernel 2: dueling MLP via V_WMMA_F32_16X16X4_F32 (fp32 tensor path, exact
// precision vs reference).  One wave per 16-row batch tile; every layer is a
// chain of K=4 WMMAs.  D-layout -> A-layout transpose between layers goes
// through a per-wave LDS tile (stride 33 floats: conflict-free).
// ---------------------------------------------------------------------------
static __device__ __forceinline__ v8f wmma4(v2f a, v2f b, v8f c) {
    return __builtin_amdgcn_wmma_f32_16x16x4_f32(false, a, false, b,
                                                 (short)0, c, false, false);
}

__global__ void __launch_bounds__(256)
mlp_kernel(const float* __restrict__ steer,
           const float* __restrict__ dist,
           const float* __restrict__ kmph,
           const float* __restrict__ W1, const float* __restrict__ b1,
           const float* __restrict__ W2, const float* __restrict__ b2,
           const float* __restrict__ W3, const float* __restrict__ b3,
           const float* __restrict__ W4, const float* __restrict__ b4,
           const float* __restrict__ Wv, const float* __restrict__ bv,
           const float* __restrict__ Wa, const float* __restrict__ ba,
           float* __restrict__ out) {
    // per-wave LDS tile: 16 rows x up-to-32 cols, stride 33 (odd -> no bank conflicts)
    __shared__ float smem[8 * 16 * 33];
    const int lane = threadIdx.x & 31;
    const int wv   = threadIdx.x >> 5;
    float* buf = smem + wv * (16 * 33);

    const int tile = blockIdx.x * 8 + wv;
    const int base = tile * 16;

    const int n    = lane & 15;   // B/C/D column, also A row (M)
    const int half = lane >> 4;
    const int kb   = half * 2;    // A/B K-pair base: {0,1} or {2,3}
    const int m    = n;           // A-layout: M = lane & 15

    // ---- layer 1: [16,3(pad 4)] x [3(pad 4),16] : one WMMA -----------------
    const int row0 = base + n;
    v2f a;
    a.x = (kb == 0) ? steer[row0] : kmph[row0];
    a.y = (kb == 0) ? dist[row0]  : 0.f;        // K=3 zero-padded to 4
    v2f bm;
    bm.x = (kb     < 3) ? W1[kb * 16 + n]       : 0.f;
    bm.y = (kb + 1 < 3) ? W1[(kb + 1) * 16 + n] : 0.f;
    v8f acc = {};
    acc = wmma4(a, bm, acc);

    // D layout: vgpr v -> (M = v + 8*half, N = n).  Fuse +b1, ReLU.
    {
        const float bias = b1[n];
        #pragma unroll
        for (int v = 0; v < 8; ++v) {
            float val = acc[v] + bias;
            buf[(v + 8 * half) * 33 + n] = val > 0.f ? val : 0.f;
        }
    }
    __syncthreads();

    // ---- layer 2: [16,16] x [16,32] : 2 N-tiles x 4 K-chunks = 8 WMMAs -----
    v8f c20 = {}, c21 = {};
    #pragma unroll
    for (int c = 0; c < 4; ++c) {
        const int k0 = 4 * c + kb;
        v2f aa; aa.x = buf[m * 33 + k0]; aa.y = buf[m * 33 + k0 + 1];
        v2f b0; b0.x = W2[k0 * 32 + n];        b0.y = W2[(k0 + 1) * 32 + n];
        v2f bB; bB.x = W2[k0 * 32 + 16 + n];   bB.y = W2[(k0 + 1) * 32 + 16 + n];
        c20 = wmma4(aa, b0, c20);
        c21 = wmma4(aa, bB, c21);
    }
    __syncthreads();
    {
        const float bias0 = b2[n], bias1 = b2[16 + n];
        #pragma unroll
        for (int v = 0; v < 8; ++v) {
            const int mm = v + 8 * half;
            float v0 = c20[v] + bias0;
            float v1 = c21[v] + bias1;
            buf[mm * 33 + n]      = v0 > 0.f ? v0 : 0.f;
            buf[mm * 33 + 16 + n] = v1 > 0.f ? v1 : 0.f;
        }
    }
    __syncthreads();

    // ---- layer 3: [16,32] x [32,32] : 2 N-tiles x 8 K-chunks = 16 WMMAs ----
    v8f c30 = {}, c31 = {};
    #pragma unroll
    for (int c = 0; c < 8; ++c) {
        const int k0 = 4 * c + kb;
        v2f aa; aa.x = buf[m * 33 + k0]; aa.y = buf[m * 33 + k0 + 1];
        v2f b0; b0.x = W3[k0 * 32 + n];        b0.y = W3[(k0 + 1) * 32 + n];
        v2f bB; bB.x = W3[k0 * 32 + 16 + n];   bB.y = W3[(k0 + 1) * 32 + 16 + n];
        c30 = wmma4(aa, b0, c30);
        c31 = wmma4(aa, bB, c31);
    }
    __syncthreads();
    {
        const float bias0 = b3[n], bias1 = b3[16 + n];
        #pragma unroll
        for (int v = 0; v < 8; ++v) {
            const int mm = v + 8 * half;
            float v0 = c30[v] + bias0;
            float v1 = c31[v] + bias1;
            buf[mm * 33 + n]      = v0 > 0.f ? v0 : 0.f;
            buf[mm * 33 + 16 + n] = v1 > 0.f ? v1 : 0.f;
        }
    }
    __syncthreads();

    // ---- layer 4: [16,32] x [32,16] : 8 WMMAs, no ReLU ---------------------
    v8f c4 = {};
    #pragma unroll
    for (int c = 0; c < 8; ++c) {
        const int k0 = 4 * c + kb;
        v2f aa; aa.x = buf[m * 33 + k0]; aa.y = buf[m * 33 + k0 + 1];
        v2f bb; bb.x = W4[k0 * 16 + n];  bb.y = W4[(k0 + 1) * 16 + n];
        c4 = wmma4(aa, bb, c4);
    }
    __syncthreads();
    {
        const float bias = b4[n];
        #pragma unroll
        for (int v = 0; v < 8; ++v)
            buf[(v + 8 * half) * 33 + n] = c4[v] + bias;   // linear
    }
    __syncthreads();

    // ---- heads fused: [16,16] x [16, {V|A0..A3|0...}] : 4 WMMAs ------------
    v8f c5 = {};
    #pragma unroll
    for (int c = 0; c < 4; ++c) {
        const int k0 = 4 * c + kb;
        v2f aa; aa.x = buf[m * 33 + k0]; aa.y = buf[m * 33 + k0 + 1];
        v2f hb;
        hb.x = (n == 0) ? Wv[k0]
             : ((n >= 1 && n <= 4) ? Wa[k0 * 4 + (n - 1)] : 0.f);
        hb.y = (n == 0) ? Wv[k0 + 1]
             : ((n >= 1 && n <= 4) ? Wa[(k0 + 1) * 4 + (n - 1)] : 0.f);
        c5 = wmma4(aa, hb, c5);
    }
    __syncthreads();
    {
        const float hbias = (n == 0) ? bv[0]
                          : ((n >= 1 && n <= 4) ? ba[n - 1] : 0.f);
        #pragma unroll
        for (int v = 0; v < 8; ++v)
            buf[(v + 8 * half) * 33 + n] = c5[v] + hbias;
    }
    __syncthreads();

    // ---- dueling combine: q = V + (A - mean(A)), 2 outputs per lane --------
    #pragma unroll
    for (int e2 = 0; e2 < 2; ++e2) {
        const int e  = lane * 2 + e2;       // 0..63 = 16 rows x 4 actions
        const int mr = e >> 2;
        const int j  = e & 3;
        const float* r = buf + mr * 33;
        const float vval = r[0];
        const float mean = 0.25f * (r[1] + r[2] + r[3] + r[4]);
        out[(base + mr) * 4 + j] = vval + r[1 + j] - mean;
    }
}

// ---------------------------------------------------------------------------
extern "C" void kernel_launch(void* const* d_in, const int* in_sizes, int n_in,
                              void* d_out, int out_size, void* d_ws, size_t ws_size,
                              hipStream_t stream) {
    const float* rgb      = (const float*)d_in[0];
    const float* distance = (const float*)d_in[1];
    const float* kmph     = (const float*)d_in[2];
    const float* laneW    = (const float*)d_in[3];
    const float* laneb    = (const float*)d_in[4];
    const float* W1 = (const float*)d_in[5];  const float* b1 = (const float*)d_in[6];
    const float* W2 = (const float*)d_in[7];  const float* b2 = (const float*)d_in[8];
    const float* W3 = (const float*)d_in[9];  const float* b3 = (const float*)d_in[10];
    const float* W4 = (const float*)d_in[11]; const float* b4 = (const float*)d_in[12];
    const float* Wv = (const float*)d_in[13]; const float* bv = (const float*)d_in[14];
    const float* Wa = (const float*)d_in[15]; const float* ba = (const float*)d_in[16];
    float* out = (float*)d_out;

    const int B = in_sizes[1];                 // 8192 (batch = |distance|)

    float* wperm = (float*)d_ws;               // [15000]
    float* steer = wperm + NPIX;               // [B]

    permute_lane_w<<<(NPIX + 255) / 256, 256, 0, stream>>>(laneW, wperm);
    steer_kernel<<<B, 256, 0, stream>>>(rgb, wperm, laneb, steer);

    const int tiles  = B / 16;                 // 512
    const int blocks = tiles / 8;              // 64 (8 waves/block, 1 tile/wave)
    mlp_kernel<<<blocks, 256, 0, stream>>>(steer, distance, kmph,
                                           W1, b1, W2, b2, W3, b3, W4, b4,
                                           Wv, bv, Wa, ba, out);
}